// CrossAttention_61830349193403
// MI455X (gfx1250) — compile-verified
//
#include <hip/hip_runtime.h>
#include <hip/hip_bf16.h>
#include <math.h>

// ---------------------------------------------------------------------------
// CDNA5 (gfx1250) wave32 implementation of the linear cross-attention block.
// Memory-bound problem => keep f32 end-to-end; V_WMMA_F32_16X16X4_F32 for all
// dense GEMMs, with A-panel staging via GLOBAL_LOAD_ASYNC_TO_LDS_B128
// (double-buffered, s_wait_asynccnt synchronized).
// ---------------------------------------------------------------------------

typedef __attribute__((ext_vector_type(2))) float v2f;
typedef __attribute__((ext_vector_type(8))) float v8f;

#define EPSL 1e-5f

__device__ __forceinline__ float wave_sum(float v) {
#pragma unroll
    for (int off = 16; off > 0; off >>= 1) v += __shfl_xor(v, off, 32);
    return v;
}

// ---------------------------------------------------------------------------
// LayerNorm over rows of 256: one wave per row, 8 elements per lane.
// ---------------------------------------------------------------------------
__global__ __launch_bounds__(256) void ln256_kernel(
    const float* __restrict__ x, const float* __restrict__ g,
    const float* __restrict__ b, float* __restrict__ out, int nrows) {
    int wave = threadIdx.x >> 5;
    int lane = threadIdx.x & 31;
    int row = blockIdx.x * 8 + wave;
    if (row >= nrows) return;
    const float* xr = x + (size_t)row * 256;
    float v[8];
    float s = 0.f;
#pragma unroll
    for (int j = 0; j < 8; ++j) { v[j] = xr[lane + 32 * j]; s += v[j]; }
    s = wave_sum(s);
    float m = s * (1.0f / 256.0f);
    float q = 0.f;
#pragma unroll
    for (int j = 0; j < 8; ++j) { v[j] -= m; q += v[j] * v[j]; }
    q = wave_sum(q);
    float rs = rsqrtf(q * (1.0f / 256.0f) + EPSL);
    float* orow = out + (size_t)row * 256;
#pragma unroll
    for (int j = 0; j < 8; ++j) {
        int c = lane + 32 * j;
        orow[c] = v[j] * rs * g[c] + b[c];
    }
}

// ---------------------------------------------------------------------------
// y = LN2( LN1(rep)*g1+b1 + x1 )*g2+b2   (rows of 256)
// ---------------------------------------------------------------------------
__global__ __launch_bounds__(256) void lnln256_kernel(
    const float* __restrict__ rep, const float* __restrict__ x1,
    const float* __restrict__ g1, const float* __restrict__ b1,
    const float* __restrict__ g2, const float* __restrict__ b2,
    float* __restrict__ y, int nrows) {
    int wave = threadIdx.x >> 5;
    int lane = threadIdx.x & 31;
    int row = blockIdx.x * 8 + wave;
    if (row >= nrows) return;
    const float* rr = rep + (size_t)row * 256;
    const float* xr = x1 + (size_t)row * 256;
    float v[8];
    float s = 0.f;
#pragma unroll
    for (int j = 0; j < 8; ++j) { v[j] = rr[lane + 32 * j]; s += v[j]; }
    s = wave_sum(s);
    float m = s * (1.0f / 256.0f);
    float q = 0.f;
#pragma unroll
    for (int j = 0; j < 8; ++j) { v[j] -= m; q += v[j] * v[j]; }
    q = wave_sum(q);
    float rs = rsqrtf(q * (1.0f / 256.0f) + EPSL);
    float s2 = 0.f;
#pragma unroll
    for (int j = 0; j < 8; ++j) {
        int c = lane + 32 * j;
        v[j] = v[j] * rs * g1[c] + b1[c] + xr[c];
        s2 += v[j];
    }
    s2 = wave_sum(s2);
    float m2 = s2 * (1.0f / 256.0f);
    float q2 = 0.f;
#pragma unroll
    for (int j = 0; j < 8; ++j) { v[j] -= m2; q2 += v[j] * v[j]; }
    q2 = wave_sum(q2);
    float rs2 = rsqrtf(q2 * (1.0f / 256.0f) + EPSL);
    float* yr = y + (size_t)row * 256;
#pragma unroll
    for (int j = 0; j < 8; ++j) {
        int c = lane + 32 * j;
        yr[c] = v[j] * rs2 * g2[c] + b2[c];
    }
}

// ---------------------------------------------------------------------------
// out = LN(a + c)   (rows of 256)
// ---------------------------------------------------------------------------
__global__ __launch_bounds__(256) void add_ln256_kernel(
    const float* __restrict__ a, const float* __restrict__ cin,
    const float* __restrict__ g, const float* __restrict__ b,
    float* __restrict__ out, int nrows) {
    int wave = threadIdx.x >> 5;
    int lane = threadIdx.x & 31;
    int row = blockIdx.x * 8 + wave;
    if (row >= nrows) return;
    const float* ar = a + (size_t)row * 256;
    const float* cr = cin + (size_t)row * 256;
    float v[8];
    float s = 0.f;
#pragma unroll
    for (int j = 0; j < 8; ++j) {
        int c = lane + 32 * j;
        v[j] = ar[c] + cr[c];
        s += v[j];
    }
    s = wave_sum(s);
    float m = s * (1.0f / 256.0f);
    float q = 0.f;
#pragma unroll
    for (int j = 0; j < 8; ++j) { v[j] -= m; q += v[j] * v[j]; }
    q = wave_sum(q);
    float rs = rsqrtf(q * (1.0f / 256.0f) + EPSL);
    float* orow = out + (size_t)row * 256;
#pragma unroll
    for (int j = 0; j < 8; ++j) {
        int c = lane + 32 * j;
        orow[c] = v[j] * rs * g[c] + b[c];
    }
}

// ---------------------------------------------------------------------------
// GEMM: C[m,n] = sum_k A[m,k]*W[n,k] + bias[n]
// A: M x K row-major, W: N x K row-major, C: M x N row-major.
//
// Block = 16 rows x 128 cols (8 waves, one 16x16 tile column per wave; the
// 16-row A strip is shared by the whole block). The A strip is staged in LDS
// in K-chunks of 64 via GLOBAL_LOAD_ASYNC_TO_LDS_B128 (one b128 per lane),
// double-buffered and synchronized with s_wait_asynccnt + barriers so the
// next chunk's DMA overlaps the current chunk's WMMAs. B (weights) stays
// direct-from-global: it is small and L2-resident.
//
// LDS A layout: row stride 68 dwords. Fragment read: lane l16 reads dwords
// 68*l16 + kk + 2*half -> bank 4*l16 + kk (+2 for upper half-wave): all 64
// banks exactly once per paired b64 read => conflict-free.
//
// WMMA fragment layouts per CDNA5 ISA 7.12.2:
//   A frag: lane = (M = lane%16), vgpr v -> K = kbase + 2*(lane>=16) + v
//   B frag: lane = (N = lane%16), vgpr v -> K = kbase + 2*(lane>=16) + v
//   D:      vgpr r -> row mbase + r + 8*(lane>=16), col nbase + lane%16
// ---------------------------------------------------------------------------
#define LDS_AROW 68
#define ASYNC_LOAD_A(ldsaddr, gptr)                                        \
    asm volatile("global_load_async_to_lds_b128 %0, %1, off"               \
                 :: "v"(ldsaddr), "v"(gptr) : "memory")

__global__ __launch_bounds__(256) void gemm_wmma_kernel(
    const float* __restrict__ A, const float* __restrict__ W,
    const float* __restrict__ bias, float* __restrict__ C,
    int M, int N, int K) {
    __shared__ float sA[2][16 * LDS_AROW];

    int tid = threadIdx.x;
    int wave = tid >> 5;
    int lane = tid & 31;
    int nbn = N >> 7;                    // column-blocks of 128
    int mblk = blockIdx.x / nbn;
    int nblk = blockIdx.x - mblk * nbn;
    int m0 = mblk << 4;
    int n0 = (nblk << 7) + (wave << 4);
    int half = lane >> 4;
    int l16 = lane & 15;

    // Staging role: thread tid copies 16B of row r at dword offset koff.
    int r = tid >> 4;
    int koff = (tid & 15) << 2;
    const float* ag = A + (size_t)(m0 + r) * K + koff;
    unsigned lds0 = (unsigned)(size_t)&sA[0][r * LDS_AROW + koff];
    unsigned lds1 = lds0 + 16 * LDS_AROW * 4;

    const float* wrow = W + (size_t)(n0 + l16) * K + 2 * half;
    v8f acc0 = {};
    v8f acc1 = {};

    ASYNC_LOAD_A(lds0, ag);              // chunk 0 -> buffer 0
    int nc = K >> 6;
    for (int c = 0; c < nc; ++c) {
        int k = c << 6;
        int cur = c & 1;
        if (c + 1 < nc) {
            // Prefetch next chunk into the other buffer (its previous data
            // was fully consumed before the trailing barrier of iter c-1).
            ASYNC_LOAD_A(cur ? lds0 : lds1, ag + k + 64);
            asm volatile("s_wait_asynccnt 0x1" ::: "memory");
        } else {
            asm volatile("s_wait_asynccnt 0x0" ::: "memory");
        }
        __syncthreads();                 // all waves' chunk-c data landed
        const float* sa = &sA[cur][0];
        const float* wk = wrow + k;
#pragma unroll
        for (int kk = 0; kk < 64; kk += 8) {
            v2f a0 = *(const v2f*)(sa + l16 * LDS_AROW + kk + 2 * half);
            v2f a1 = *(const v2f*)(sa + l16 * LDS_AROW + kk + 4 + 2 * half);
            v2f b0 = *(const v2f*)(wk + kk);
            v2f b1 = *(const v2f*)(wk + kk + 4);
            acc0 = __builtin_amdgcn_wmma_f32_16x16x4_f32(
                false, a0, false, b0, (short)0, acc0, false, false);
            acc1 = __builtin_amdgcn_wmma_f32_16x16x4_f32(
                false, a1, false, b1, (short)0, acc1, false, false);
        }
        __syncthreads();                 // done reading buf cur before reuse
    }

    float bn = bias[n0 + l16];
    float* crow = C + (size_t)(m0 + 8 * half) * N + n0 + l16;
#pragma unroll
    for (int rr2 = 0; rr2 < 8; ++rr2) {
        crow[(size_t)rr2 * N] = acc0[rr2] + acc1[rr2] + bn;
    }
}

// ---------------------------------------------------------------------------
// Linear attention core. One workgroup per (b, head).
// n1/n2 are (B, N=4096, C=256); head channels are n[..., h*32 : h*32+32].
//   Z[d]      = sum_n exp(k[d,n])
//   Mctx[d,v] = sum_n exp(k[d,n]) * k[v,n]        (v uses raw values)
//   ctx[d,v]  = Mctx[d,v] / Z[d]
//   att[v,n]  = sum_d ctx[d,v] * softmax_d(q[:,n])[d]
// Writes aggT[b, n, h*32+v]  (so reproj is a plain row-major GEMM).
// f32 exp is overflow-safe here (post-LN values are ~N(0,1)), so the max
// subtraction in softmax can be dropped without changing the math.
// ---------------------------------------------------------------------------
#define ATTN_TN 64
__global__ __launch_bounds__(256) void linattn_kernel(
    const float* __restrict__ n1, const float* __restrict__ n2,
    float* __restrict__ aggT) {
    const int Ntok = 4096, Cc = 256, dk = 32;
    int hd = blockIdx.x;  // head 0..7
    int b = blockIdx.y;   // batch 0..7
    int tid = threadIdx.x;
    const float* kb = n2 + (size_t)b * Ntok * Cc + hd * dk;
    const float* qb = n1 + (size_t)b * Ntok * Cc + hd * dk;

    __shared__ float kbuf[ATTN_TN * 32];
    __shared__ float ebuf[ATTN_TN * 32];
    __shared__ float zbuf[32];
    __shared__ float ctx[32 * 32];

    int v = tid & 31;        // value-channel owned by this thread
    int dbase = tid >> 5;    // d = dbase + 8*i, i in 0..3
    float acc[4] = {0.f, 0.f, 0.f, 0.f};
    float zacc[4] = {0.f, 0.f, 0.f, 0.f};

    for (int nb = 0; nb < Ntok; nb += ATTN_TN) {
        __syncthreads();
        for (int i = tid; i < ATTN_TN * 32; i += 256) {
            int t = i >> 5, d = i & 31;
            float kv = kb[(size_t)(nb + t) * Cc + d];
            kbuf[i] = kv;
            ebuf[i] = expf(kv);
        }
        __syncthreads();
        for (int t = 0; t < ATTN_TN; ++t) {
            float kv = kbuf[t * 32 + v];
#pragma unroll
            for (int i = 0; i < 4; ++i) {
                float e = ebuf[t * 32 + dbase + 8 * i];
                acc[i] += e * kv;
                zacc[i] += e;   // identical across v; only v==0 publishes
            }
        }
    }
    __syncthreads();
    if (v == 0) {
#pragma unroll
        for (int i = 0; i < 4; ++i) zbuf[dbase + 8 * i] = zacc[i];
    }
    __syncthreads();
#pragma unroll
    for (int i = 0; i < 4; ++i) {
        int d = dbase + 8 * i;
        ctx[d * 32 + v] = acc[i] / zbuf[d];
    }
    __syncthreads();

    // Phase 2: per-token channel softmax of q and 32x32 apply.
    float e[32], att[32];
    for (int j = 0; j < 16; ++j) {
        int n = j * 256 + tid;
        const float* qp = qb + (size_t)n * Cc;
        float s = 0.f;
#pragma unroll
        for (int d = 0; d < 32; ++d) { e[d] = expf(qp[d]); s += e[d]; }
        float inv = 1.0f / s;
#pragma unroll
        for (int vv = 0; vv < 32; ++vv) att[vv] = 0.f;
        for (int d = 0; d < 32; ++d) {
            float ed = e[d];
#pragma unroll
            for (int vv = 0; vv < 32; ++vv) att[vv] += ed * ctx[d * 32 + vv];
        }
        float* op = aggT + ((size_t)b * Ntok + n) * Cc + hd * dk;
#pragma unroll
        for (int vv = 0; vv < 32; ++vv) op[vv] = att[vv] * inv;
    }
}

// ---------------------------------------------------------------------------
// Fused depthwise 3x3 conv (SAME) + LN over 512 channels + exact GELU.
// One workgroup per (b, y, x); 256 threads x 2 channels each.
// h1 layout: (B, 64, 64, 512) NHWC.  dw_w: (512,1,3,3).
// ---------------------------------------------------------------------------
__device__ __forceinline__ float block_sum_256(float vsum, float* sred) {
    int lane = threadIdx.x & 31, wave = threadIdx.x >> 5;
    float w = wave_sum(vsum);
    __syncthreads();
    if (lane == 0) sred[wave] = w;
    __syncthreads();
    float t = sred[0];
#pragma unroll
    for (int i = 1; i < 8; ++i) t += sred[i];
    return t;
}

__global__ __launch_bounds__(256) void dwconv_ln_gelu_kernel(
    const float* __restrict__ h1, const float* __restrict__ w,
    const float* __restrict__ wb, const float* __restrict__ g,
    const float* __restrict__ bb, float* __restrict__ out) {
    const int C2 = 512;
    int sp = blockIdx.x;          // y*64 + x
    int b = blockIdx.y;
    int y = sp >> 6, x = sp & 63;
    int tid = threadIdx.x;
    __shared__ float sred[8];

    float vals[2];
#pragma unroll
    for (int j = 0; j < 2; ++j) {
        int c = tid + j * 256;
        float a = wb[c];
#pragma unroll
        for (int dy = -1; dy <= 1; ++dy) {
            int yy = y + dy;
            if ((unsigned)yy >= 64u) continue;
#pragma unroll
            for (int dx = -1; dx <= 1; ++dx) {
                int xx = x + dx;
                if ((unsigned)xx >= 64u) continue;
                a += h1[(((size_t)b * 64 + yy) * 64 + xx) * C2 + c] *
                     w[c * 9 + (dy + 1) * 3 + (dx + 1)];
            }
        }
        vals[j] = a;
    }
    float s = block_sum_256(vals[0] + vals[1], sred);
    float m = s * (1.0f / 512.0f);
    float d0 = vals[0] - m, d1 = vals[1] - m;
    float q = block_sum_256(d0 * d0 + d1 * d1, sred);
    float rs = rsqrtf(q * (1.0f / 512.0f) + EPSL);
    size_t base = (((size_t)b * 64 + y) * 64 + x) * C2;
    float dv[2] = {d0, d1};
#pragma unroll
    for (int j = 0; j < 2; ++j) {
        int c = tid + j * 256;
        float t = dv[j] * rs * g[c] + bb[c];
        out[base + c] = 0.5f * t * (1.0f + erff(t * 0.70710678118654752f));
    }
}

// ---------------------------------------------------------------------------
// Host-side orchestration.
// ---------------------------------------------------------------------------
extern "C" void kernel_launch(void* const* d_in, const int* in_sizes, int n_in,
                              void* d_out, int out_size, void* d_ws, size_t ws_size,
                              hipStream_t stream) {
    (void)in_sizes; (void)n_in; (void)out_size; (void)ws_size;
    const float* x1       = (const float*)d_in[0];
    const float* x2       = (const float*)d_in[1];
    const float* norm1_g  = (const float*)d_in[2];
    const float* norm1_b  = (const float*)d_in[3];
    const float* lin_w    = (const float*)d_in[4];
    const float* lin_b    = (const float*)d_in[5];
    const float* reproj_w = (const float*)d_in[6];
    const float* reproj_b = (const float*)d_in[7];
    const float* attn_g   = (const float*)d_in[8];
    const float* attn_b   = (const float*)d_in[9];
    const float* norm2_g  = (const float*)d_in[10];
    const float* norm2_b  = (const float*)d_in[11];
    const float* fc1_w    = (const float*)d_in[12];
    const float* fc1_b    = (const float*)d_in[13];
    const float* dw_w     = (const float*)d_in[14];
    const float* dw_b     = (const float*)d_in[15];
    const float* ffn1_g   = (const float*)d_in[16];
    const float* ffn1_b   = (const float*)d_in[17];
    const float* fc2_w    = (const float*)d_in[18];
    const float* fc2_b    = (const float*)d_in[19];
    const float* ffn3_g   = (const float*)d_in[20];
    const float* ffn3_b   = (const float*)d_in[21];
    float* outp = (float*)d_out;

    const int Mrows = 8 * 64 * 64;          // 32768
    const size_t R = (size_t)Mrows * 256;   // 8M floats = 32MB
    float* ws = (float*)d_ws;
    float* P0 = ws;          // xn1 -> aggT -> (ax low half)
    float* P1 = ws + R;      // xn2 -> rep  -> (ax high half)
    float* P2 = ws + 2 * R;  // n1  -> y
    float* P3 = ws + 3 * R;  // n2  -> fc2 out
    float* P4 = ws + 4 * R;  // h1 (16M floats = 64MB)
    float* AX = P0;          // 16M floats spanning P0..P1

    dim3 blk(256);

    // 1-2: LN(x1), LN(x2)
    ln256_kernel<<<Mrows / 8, blk, 0, stream>>>(x1, norm1_g, norm1_b, P0, Mrows);
    ln256_kernel<<<Mrows / 8, blk, 0, stream>>>(x2, norm1_g, norm1_b, P1, Mrows);

    // 3-4: n1 = xn1 @ lin_w^T + b ; n2 = xn2 @ lin_w^T + b
    int blocks_256 = (Mrows / 16) * (256 / 128);  // 4096 (16-row x 128-col blocks)
    gemm_wmma_kernel<<<blocks_256, blk, 0, stream>>>(P0, lin_w, lin_b, P2, Mrows, 256, 256);
    gemm_wmma_kernel<<<blocks_256, blk, 0, stream>>>(P1, lin_w, lin_b, P3, Mrows, 256, 256);

    // 5: linear attention -> aggT (B,N,C)
    linattn_kernel<<<dim3(8, 8), blk, 0, stream>>>(P2, P3, P0);

    // 6: rep = aggT @ reproj_w^T + b
    gemm_wmma_kernel<<<blocks_256, blk, 0, stream>>>(P0, reproj_w, reproj_b, P1, Mrows, 256, 256);

    // 7: y = LN2( LN1(rep)+x1 )
    lnln256_kernel<<<Mrows / 8, blk, 0, stream>>>(P1, x1, attn_g, attn_b,
                                                  norm2_g, norm2_b, P2, Mrows);

    // 8: h1 = y @ fc1_w^T + b  (N=512)
    int blocks_512 = (Mrows / 16) * (512 / 128);  // 8192
    gemm_wmma_kernel<<<blocks_512, blk, 0, stream>>>(P2, fc1_w, fc1_b, P4, Mrows, 512, 256);

    // 9: ax = gelu(LN512(dwconv3x3(h1)))
    dwconv_ln_gelu_kernel<<<dim3(4096, 8), blk, 0, stream>>>(P4, dw_w, dw_b,
                                                             ffn1_g, ffn1_b, AX);

    // 10: out = ax @ fc2_w^T + b  (K=512)
    gemm_wmma_kernel<<<blocks_256, blk, 0, stream>>>(AX, fc2_w, fc2_b, P3, Mrows, 256, 512);

    // 11: final = LN(out + y)
    add_ln256_kernel<<<Mrows / 8, blk, 0, stream>>>(P3, P2, ffn3_g, ffn3_b, outp, Mrows);
}